// FeatureNet_12661563589187
// MI455X (gfx1250) — compile-verified
//
#include <hip/hip_runtime.h>
#include <hip/hip_bf16.h>

typedef __attribute__((ext_vector_type(2))) float v2f;
typedef __attribute__((ext_vector_type(8))) float v8f;

#define NPTS   700000      // V*P = 20000*35
#define NVOX   20000
#define PPV    35
#define GD     10
#define GH     200
#define GW     176
#define GHW    (GH*GW)     // 35200
#define GDHW   (GD*GH*GW)  // 352000
#define BN_EPS 1e-5f

#if __has_builtin(__builtin_amdgcn_global_load_async_to_lds_b128) && \
    __has_builtin(__builtin_amdgcn_s_wait_asynccnt)
#define HAS_ASYNC_LDS 1
#endif

typedef int v4i __attribute__((vector_size(16)));
typedef __attribute__((address_space(1))) v4i v4i_g;   // global int4
typedef __attribute__((address_space(3))) v4i v4i_l;   // LDS int4

__device__ __forceinline__ void copy16_g2l(const float* g, float* l) {
#ifdef HAS_ASYNC_LDS
    __builtin_amdgcn_global_load_async_to_lds_b128(
        (v4i_g*)(unsigned long long)g, (v4i_l*)l, 0, 0);
#else
    *(float4*)l = *(const float4*)g;
#endif
}

// ---------------------------------------------------------------- zero fill
__global__ void zero_kernel(float* __restrict__ p, size_t n) {
    size_t i = (size_t)blockIdx.x * blockDim.x + threadIdx.x;
    size_t stride = (size_t)gridDim.x * blockDim.x;
    for (; i < n; i += stride) p[i] = 0.0f;
}

// ---------------------------------------------------------------- keep mask
__global__ __launch_bounds__(256) void keep_kernel(const float* __restrict__ x,
                                                   float* __restrict__ keep, int npts) {
    int i = blockIdx.x * 256 + threadIdx.x;
    if (i >= npts) return;
    float s = 0.0f;
#pragma unroll
    for (int c = 0; c < 7; ++c) s += x[i * 7 + c];
    keep[i] = (s == 0.0f) ? 0.0f : 1.0f;
}

// ------------------------------------------------- VFE1 linear (K=7, scalar)
__global__ __launch_bounds__(256) void vfe1_linear_kernel(
    const float* __restrict__ x, const float* __restrict__ W1,
    const float* __restrict__ b1, float* __restrict__ out,
    float* __restrict__ gsum, float* __restrict__ gsq, int npts)
{
    __shared__ float sSum[16];
    __shared__ float sSq[16];
    int tid = threadIdx.x;
    if (tid < 16) { sSum[tid] = 0.0f; sSq[tid] = 0.0f; }
    __syncthreads();

    int i = blockIdx.x * 256 + tid;
    if (i < npts) {
        float xi[7];
#pragma unroll
        for (int c = 0; c < 7; ++c) xi[c] = x[i * 7 + c];
#pragma unroll
        for (int u = 0; u < 16; ++u) {
            float acc = b1[u];
#pragma unroll
            for (int c = 0; c < 7; ++c) acc = fmaf(xi[c], W1[c * 16 + u], acc);
            out[i * 16 + u] = acc;
            atomicAdd(&sSum[u], acc);        // ds_add_f32
            atomicAdd(&sSq[u],  acc * acc);
        }
    }
    __syncthreads();
    if (tid < 16) { atomicAdd(&gsum[tid], sSum[tid]); atomicAdd(&gsq[tid], sSq[tid]); }
}

// ----------------------------------------------- BN finalize (scale / shift)
__global__ void bn_finalize_kernel(const float* __restrict__ sum, const float* __restrict__ sq,
                                   const float* __restrict__ gamma, const float* __restrict__ beta,
                                   float* __restrict__ scale, float* __restrict__ shift,
                                   int nchan, float inv_count) {
    int c = blockIdx.x * blockDim.x + threadIdx.x;
    if (c >= nchan) return;
    float mean = sum[c] * inv_count;
    float var  = sq[c] * inv_count - mean * mean;
    float sc   = gamma[c] * rsqrtf(var + BN_EPS);
    scale[c] = sc;
    shift[c] = beta[c] - mean * sc;
}

// --------------- apply BN+ReLU, voxel-max over P, concat [h, max], mask keep
template <int U>
__global__ __launch_bounds__(256) void vfe_apply_kernel(
    const float* __restrict__ pre, const float* __restrict__ scale,
    const float* __restrict__ shift, const float* __restrict__ keep,
    float* __restrict__ out, int nvox)
{
    constexpr int VPB = 256 / U;
    int tid = threadIdx.x;
    int c = tid % U;
    int v = blockIdx.x * VPB + tid / U;
    if (v >= nvox) return;

    float sc = scale[c], sh = shift[c];
    const float* p0 = pre + (size_t)v * PPV * U + c;
    float m = 0.0f;                                  // ReLU outputs are >= 0
#pragma unroll
    for (int p = 0; p < PPV; ++p)
        m = fmaxf(m, fmaxf(fmaf(p0[p * U], sc, sh), 0.0f));

    float* o0 = out + (size_t)v * PPV * (2 * U) + c;
#pragma unroll
    for (int p = 0; p < PPV; ++p) {
        float k = keep[v * PPV + p];
        float y = fmaxf(fmaf(p0[p * U], sc, sh), 0.0f);
        o0[p * 2 * U]     = y * k;
        o0[p * 2 * U + U] = m * k;
    }
}

// --------------------- fp32 WMMA GEMM (M=NPTS x K=CIN x N=COUT) + BN stats
// Per block step: one contiguous group of WM 16-row A tiles is async-staged
// into LDS (double-buffered, ASYNCcnt) and shared by all 8 waves. Each wave
// owns a 16-wide N tile, keeps all B fragments in registers, and accumulates
// with v_wmma_f32_16x16x4_f32 reading A fragments from LDS (ds_load_2addr_b64,
// bank-conflict free thanks to +4 dword row padding).
template <int CIN, int COUT, int STEPS>
__global__ __launch_bounds__(256) void gemm_bn_stats_kernel(
    const float* __restrict__ A, const float* __restrict__ Wm,
    const float* __restrict__ bias, float* __restrict__ Out,
    float* __restrict__ gsum, float* __restrict__ gsq, int mtiles)
{
    constexpr int NT      = COUT / 16;           // N tiles (waves per m-slot)
    constexpr int WM      = (256 / 32) / NT;     // m-tiles staged per step
    constexpr int LROW    = CIN + 4;             // padded LDS row (dwords)
    constexpr int TILE_F4 = WM * 16 * CIN / 4;   // float4 chunks per step
    constexpr int ISS     = TILE_F4 / 256;       // async issues per thread
    constexpr int KSTEPS  = CIN / 4;

    __shared__ __align__(16) float sA[2][WM * 16 * LROW];
    __shared__ float sSum[COUT];
    __shared__ float sSq[COUT];

    int tid = threadIdx.x;
    for (int i = tid; i < COUT; i += 256) { sSum[i] = 0.0f; sSq[i] = 0.0f; }

    int wave    = tid >> 5;
    int lane    = tid & 31;
    int nt      = wave % NT;
    int slot    = wave / NT;
    int lane_lo = lane & 15;
    int khalf   = (lane >> 4) * 2;    // lanes 0-15: K 0,1 ; lanes 16-31: K 2,3

    // B fragments (register-resident, reused for every m-tile)
    int ncol = nt * 16 + lane_lo;
    v2f bfrag[KSTEPS];
#pragma unroll
    for (int ks = 0; ks < KSTEPS; ++ks) {
        int krow = ks * 4 + khalf;
        bfrag[ks].x = Wm[krow * COUT + ncol];
        bfrag[ks].y = Wm[(krow + 1) * COUT + ncol];
    }
    float bv = bias[ncol];

    auto stage = [&](int step, int buf) {
        const float* gbase = A + (size_t)(blockIdx.x * STEPS + step) * (WM * 16 * CIN);
#pragma unroll
        for (int r = 0; r < ISS; ++r) {
            int f4  = tid + r * 256;             // flat float4 index in tile group
            int t16 = f4 / (16 * CIN / 4);       // which m-tile slot
            int rem = f4 % (16 * CIN / 4);
            int row = rem / (CIN / 4);
            int cg  = rem % (CIN / 4);
            const float* g = gbase + (size_t)f4 * 4;
            float* l = &sA[buf][(t16 * 16 + row) * LROW + cg * 4];
            __builtin_prefetch(g + WM * 16 * CIN, 0, 0);  // global_prefetch_b8
            copy16_g2l(g, l);
        }
    };

    stage(0, 0);

    float s = 0.0f, sq = 0.0f;
#pragma unroll 1
    for (int i = 0; i < STEPS; ++i) {
        if (i + 1 < STEPS) {
            stage(i + 1, (i + 1) & 1);
#ifdef HAS_ASYNC_LDS
            __builtin_amdgcn_s_wait_asynccnt(ISS);   // previous tile done
#endif
        } else {
#ifdef HAS_ASYNC_LDS
            __builtin_amdgcn_s_wait_asynccnt(0);
#endif
        }
        __syncthreads();

        int mt = (blockIdx.x * STEPS + i) * WM + slot;
        if (mt < mtiles) {
            const float* As = &sA[i & 1][slot * 16 * LROW];
            v8f acc = {};
#pragma unroll
            for (int ks = 0; ks < KSTEPS; ++ks) {
                v2f a = *(const v2f*)&As[lane_lo * LROW + ks * 4 + khalf];
                acc = __builtin_amdgcn_wmma_f32_16x16x4_f32(
                    false, a, false, bfrag[ks], (short)0, acc, false, false);
            }
            // C/D layout: VGPR r -> row (r + (lane>>4)*8), col = lane&15
            // Single 64-bit base per tile; r*COUT folds into store imm offsets.
            float* obase = Out + (size_t)mt * (16 * COUT)
                               + (lane >> 4) * (8 * COUT) + ncol;
#pragma unroll
            for (int r = 0; r < 8; ++r) {
                float v = acc[r] + bv;
                obase[r * COUT] = v;
                s += v; sq += v * v;
            }
        }
        __syncthreads();
    }

    atomicAdd(&sSum[ncol], s);   // ds_add_f32; two lanes per channel
    atomicAdd(&sSq[ncol],  sq);
    __syncthreads();
    for (int i = tid; i < COUT; i += 256) {
        atomicAdd(&gsum[i], sSum[i]);
        atomicAdd(&gsq[i],  sSq[i]);
    }
}

// ------------- final: BN+ReLU, max over P (no keep), scatter-add into grid
__global__ __launch_bounds__(128) void final_scatter_kernel(
    const float* __restrict__ pre, const float* __restrict__ scale,
    const float* __restrict__ shift, const int* __restrict__ coords,
    float* __restrict__ grid)
{
    int c = threadIdx.x;          // 0..127
    int v = blockIdx.x;           // voxel
    float sc = scale[c], sh = shift[c];
    const float* p0 = pre + (size_t)v * PPV * 128 + c;
    float m = 0.0f;
#pragma unroll
    for (int p = 0; p < PPV; ++p)
        m = fmaxf(m, fmaxf(fmaf(p0[p * 128], sc, sh), 0.0f));
    int cz = coords[v * 3 + 0];
    int cy = coords[v * 3 + 1];
    int cx = coords[v * 3 + 2];
    atomicAdd(&grid[(size_t)c * GDHW + cz * GHW + cy * GW + cx], m);
}

// ---------------------------------------------------------------------------
extern "C" void kernel_launch(void* const* d_in, const int* in_sizes, int n_in,
                              void* d_out, int out_size, void* d_ws, size_t ws_size,
                              hipStream_t stream) {
    const float* x     = (const float*)d_in[0];
    const int*   coord = (const int*)d_in[1];
    const float* W1 = (const float*)d_in[2];
    const float* b1 = (const float*)d_in[3];
    const float* g1 = (const float*)d_in[4];
    const float* be1= (const float*)d_in[5];
    const float* W2 = (const float*)d_in[6];
    const float* b2 = (const float*)d_in[7];
    const float* g2 = (const float*)d_in[8];
    const float* be2= (const float*)d_in[9];
    const float* Wd = (const float*)d_in[10];
    const float* bd = (const float*)d_in[11];
    const float* gd = (const float*)d_in[12];
    const float* bed= (const float*)d_in[13];

    float* ws = (float*)d_ws;
    // stats block (1024 floats)
    float* s1sum = ws +   0; float* s1sq = ws +  16; float* s1sc = ws +  32; float* s1sh = ws +  48;
    float* s2sum = ws +  64; float* s2sq = ws + 128; float* s2sc = ws + 192; float* s2sh = ws + 256;
    float* s3sum = ws + 320; float* s3sq = ws + 448; float* s3sc = ws + 576; float* s3sh = ws + 704;
    // big buffers
    float* keep  = ws + 1024;
    float* h1pre = keep  + (size_t)NPTS;         // [NPTS,16]
    float* h2in  = h1pre + (size_t)NPTS * 16;    // [NPTS,32]
    float* h2pre = h2in  + (size_t)NPTS * 32;    // [NPTS,64]
    float* h3in  = h2pre + (size_t)NPTS * 64;    // [NPTS,128]
    float* h3pre = h3in  + (size_t)NPTS * 128;   // [NPTS,128]

    float* grid = (float*)d_out;
    const float invN = 1.0f / (float)NPTS;
    const int MTILES = NPTS / 16;                // 43750 exactly

    // zero outputs + stats accumulators
    zero_kernel<<<2048, 256, 0, stream>>>(grid, (size_t)out_size);
    zero_kernel<<<4, 256, 0, stream>>>(ws, 1024);

    // masks
    keep_kernel<<<(NPTS + 255) / 256, 256, 0, stream>>>(x, keep, NPTS);

    // --- VFE layer 1 (7 -> 16, concat -> 32) ---
    vfe1_linear_kernel<<<(NPTS + 255) / 256, 256, 0, stream>>>(x, W1, b1, h1pre, s1sum, s1sq, NPTS);
    bn_finalize_kernel<<<1, 128, 0, stream>>>(s1sum, s1sq, g1, be1, s1sc, s1sh, 16, invN);
    vfe_apply_kernel<16><<<(NVOX + 15) / 16, 256, 0, stream>>>(h1pre, s1sc, s1sh, keep, h2in, NVOX);

    // --- VFE layer 2 (32 -> 64, concat -> 128) --- WMMA fp32 + async LDS
    {
        constexpr int STEPS = 8, WM = 2;                 // 16 m-tiles per block
        int blocks = (MTILES + STEPS * WM - 1) / (STEPS * WM);
        gemm_bn_stats_kernel<32, 64, STEPS><<<blocks, 256, 0, stream>>>(
            h2in, W2, b2, h2pre, s2sum, s2sq, MTILES);
    }
    bn_finalize_kernel<<<1, 128, 0, stream>>>(s2sum, s2sq, g2, be2, s2sc, s2sh, 64, invN);
    vfe_apply_kernel<64><<<(NVOX + 3) / 4, 256, 0, stream>>>(h2pre, s2sc, s2sh, keep, h3in, NVOX);

    // --- dense (128 -> 128) --- WMMA fp32 + async LDS
    {
        constexpr int STEPS = 8, WM = 1;                 // 8 m-tiles per block
        int blocks = (MTILES + STEPS * WM - 1) / (STEPS * WM);
        gemm_bn_stats_kernel<128, 128, STEPS><<<blocks, 256, 0, stream>>>(
            h3in, Wd, bd, h3pre, s3sum, s3sq, MTILES);
    }
    bn_finalize_kernel<<<1, 128, 0, stream>>>(s3sum, s3sq, gd, bed, s3sc, s3sh, 128, invN);

    // --- max over points + scatter into voxel grid ---
    final_scatter_kernel<<<NVOX, 128, 0, stream>>>(h3pre, s3sc, s3sh, coord, grid);
}